// LocalGrouper_3118146257059
// MI455X (gfx1250) — compile-verified
//
#include <hip/hip_runtime.h>
#include <stdint.h>

// Problem constants (fixed by the reference harness).
#define B_   8
#define N_   8192
#define C_   64
#define G_   2048
#define K_   24
#define CH_  (C_ + 3)       // 67  grouped channels
#define OC_  (2 * C_ + 3)   // 131 output channels
#define CHUNK_ 512          // points per KNN chunk staged in LDS
#define NCHUNK_ (N_ / CHUNK_)

typedef __attribute__((ext_vector_type(2))) float v2f;
typedef __attribute__((ext_vector_type(8))) float v8f;

// ---------------------------------------------------------------------------
// Kernel 1: farthest point sampling. One workgroup per batch, 1024 threads,
// each thread owns 8 points fully in registers. Two-level argmax per step.
// Also writes new_xyz (= xyz[fps_idx]) directly to d_out.
// ---------------------------------------------------------------------------
__global__ __launch_bounds__(1024) void fps_kernel(const float* __restrict__ xyz,
                                                   float* __restrict__ new_xyz,
                                                   int* __restrict__ fps_idx) {
  const int b = blockIdx.x;
  const int t = threadIdx.x;
  const float* x = xyz + (size_t)b * N_ * 3;

  __shared__ float cent[3];
  __shared__ float sd[32];
  __shared__ int   si[32];
  __shared__ int   sfar;

  float px[8], py[8], pz[8], dist[8];
#pragma unroll
  for (int i = 0; i < 8; ++i) {
    const int p = t + i * 1024;
    px[i] = x[p * 3 + 0];
    py[i] = x[p * 3 + 1];
    pz[i] = x[p * 3 + 2];
    dist[i] = 1e10f;
  }

  int far = 0;
  for (int g = 0; g < G_; ++g) {
    if (t == 0) {
      fps_idx[b * G_ + g] = far;
      const float cx = x[far * 3 + 0];
      const float cy = x[far * 3 + 1];
      const float cz = x[far * 3 + 2];
      cent[0] = cx; cent[1] = cy; cent[2] = cz;
      float* o = new_xyz + ((size_t)b * G_ + g) * 3;
      o[0] = cx; o[1] = cy; o[2] = cz;
    }
    __syncthreads();
    const float cx = cent[0], cy = cent[1], cz = cent[2];

    float bestd = -1.0f;
    int   besti = 0;
#pragma unroll
    for (int i = 0; i < 8; ++i) {
      const float dx = px[i] - cx, dy = py[i] - cy, dz = pz[i] - cz;
      const float d = dx * dx + dy * dy + dz * dz;
      dist[i] = fminf(dist[i], d);
      const int p = t + i * 1024;
      if (dist[i] > bestd || (dist[i] == bestd && p < besti)) { bestd = dist[i]; besti = p; }
    }
    // wave32 reduction (argmax, tie -> lower index)
#pragma unroll
    for (int off = 16; off > 0; off >>= 1) {
      const float od = __shfl_down(bestd, (unsigned)off, 32);
      const int   oi = __shfl_down(besti, (unsigned)off, 32);
      if (od > bestd || (od == bestd && oi < besti)) { bestd = od; besti = oi; }
    }
    if ((t & 31) == 0) { sd[t >> 5] = bestd; si[t >> 5] = besti; }
    __syncthreads();
    if (t < 32) {
      bestd = sd[t];
      besti = si[t];
#pragma unroll
      for (int off = 16; off > 0; off >>= 1) {
        const float od = __shfl_down(bestd, (unsigned)off, 32);
        const int   oi = __shfl_down(besti, (unsigned)off, 32);
        if (od > bestd || (od == bestd && oi < besti)) { bestd = od; besti = oi; }
      }
      if (t == 0) sfar = besti;
    }
    __syncthreads();
    far = sfar;
  }
}

// ---------------------------------------------------------------------------
// Async global->LDS staging of one 512-point xyz chunk (1536 dwords).
// 512 threads, 3 dwords each, thread-linear (coalesced). Tracked by ASYNCcnt.
// VDST operand = VGPR holding the wave-relative LDS byte address (low 32 bits
// of the generic shared-pointer); "off" = no SADDR (GV addressing).
// ---------------------------------------------------------------------------
__device__ __forceinline__ void stage_chunk(const float* __restrict__ xsrc,
                                            int chunkBase, uint32_t ldsBase, int t) {
#pragma unroll
  for (int k = 0; k < 3; ++k) {
    const int dw = t + k * 512;  // 0..1535
    const float* g = xsrc + (size_t)chunkBase * 3 + dw;
    const uint32_t l = ldsBase + (uint32_t)dw * 4u;
    asm volatile("global_load_async_to_lds_b32 %0, %1, off"
                 :: "v"(l), "v"(g)
                 : "memory");
    if (k == 0) __builtin_prefetch(g + 2 * CHUNK_ * 3, 0, 1);  // chunk+2 hint
  }
}

__device__ __forceinline__ void wait_async0() {
  asm volatile("s_wait_asynccnt 0x0" ::: "memory");
}

// ---------------------------------------------------------------------------
// Kernel 2: KNN. 16 centers per workgroup (16 waves x 32 lanes).
// Point xyz chunks are double-buffered in LDS via async global->LDS copies
// (software pipelined). Distances via V_WMMA_F32_16X16X4_F32 with augmented
// 4-vectors:
//   A row m = (cx, cy, cz, |c|^2), B col n = (-2px, -2py, -2pz, 1)
//   D[m][n] = |c|^2 - 2 c.p  ;  each lane then adds |p_n|^2 (its column).
// Tiles land in LDS [16 centers x CHUNK points]; wave w selects top-24 for
// center w via register-resident sorted lists + wave min-pop merge.
// ---------------------------------------------------------------------------
__global__ __launch_bounds__(512) void knn_kernel(const float* __restrict__ xyz,
                                                  const float* __restrict__ new_xyz,
                                                  int* __restrict__ topk) {
  __shared__ float dchunk[16 * CHUNK_];     // 32 KB distance tile
  __shared__ float sxyz[2][CHUNK_ * 3];     // 2 x 6 KB staged point coords

  const int t     = threadIdx.x;
  const int b     = blockIdx.x / (G_ / 16);
  const int gBase = (blockIdx.x % (G_ / 16)) * 16;
  const int lane  = t & 31;
  const int wv    = t >> 5;
  const int col   = lane & 15;
  const bool hi   = lane >= 16;

  const float* x = xyz + (size_t)b * N_ * 3;
  const uint32_t ldsX0 = (uint32_t)(uintptr_t)&sxyz[0][0];
  const uint32_t ldsX1 = (uint32_t)(uintptr_t)&sxyz[1][0];

  // A fragment: centers for this block (center 'col' in lanes col / col+16).
  const float* cp = new_xyz + ((size_t)(b * G_ + gBase + col)) * 3;
  const float cx = cp[0], cy = cp[1], cz = cp[2];
  const float cn = cx * cx + cy * cy + cz * cz;
  v2f afrag;
  afrag.x = hi ? cz : cx;
  afrag.y = hi ? cn : cy;

  // Per-lane sorted top-24 (ascending). Fully static indexing -> registers.
  float bd[K_];
  int   bi[K_];
#pragma unroll
  for (int j = 0; j < K_; ++j) { bd[j] = 1e30f; bi[j] = 0; }

  // Prologue: stage chunk 0.
  stage_chunk(x, 0, ldsX0, t);
  wait_async0();
  __syncthreads();

  for (int ch = 0; ch < NCHUNK_; ++ch) {
    const int chunkBase = ch * CHUNK_;
    const float* xb = sxyz[ch & 1];

    // ---- compute phase: wave wv produces 2 tiles of 16 points from LDS ----
#pragma unroll
    for (int t2 = 0; t2 < 2; ++t2) {
      const int pos = wv * 32 + t2 * 16 + col;  // position within chunk
      const float pxv = xb[pos * 3 + 0];
      const float pyv = xb[pos * 3 + 1];
      const float pzv = xb[pos * 3 + 2];
      const float pn  = pxv * pxv + pyv * pyv + pzv * pzv;

      v2f bfrag;
      bfrag.x = hi ? (-2.0f * pzv) : (-2.0f * pxv);
      bfrag.y = hi ? 1.0f : (-2.0f * pyv);

      v8f acc = {};
      acc = __builtin_amdgcn_wmma_f32_16x16x4_f32(
          /*neg_a=*/false, afrag, /*neg_b=*/false, bfrag,
          /*c_mod=*/(short)0, acc, /*reuse_a=*/false, /*reuse_b=*/false);

#pragma unroll
      for (int r = 0; r < 8; ++r) {
        const int m = r + (hi ? 8 : 0);          // center row of this element
        dchunk[m * CHUNK_ + pos] = acc[r] + pn;
      }
    }

    // Pipeline: kick off async staging of the next chunk into the other buf.
    if (ch + 1 < NCHUNK_) {
      stage_chunk(x, chunkBase + CHUNK_, (ch & 1) ? ldsX0 : ldsX1, t);
    }
    __syncthreads();  // dchunk tiles visible

    // ---- selection phase: wave wv scans its center's CHUNK_ distances ----
#pragma unroll 1
    for (int i = 0; i < CHUNK_ / 32; ++i) {
      const int pos = i * 32 + lane;
      const float d = dchunk[wv * CHUNK_ + pos];
      if (d < bd[K_ - 1]) {
        bd[K_ - 1] = d;
        bi[K_ - 1] = chunkBase + pos;
#pragma unroll
        for (int j = K_ - 1; j > 0; --j) {
          if (bd[j] < bd[j - 1]) {
            const float td = bd[j]; bd[j] = bd[j - 1]; bd[j - 1] = td;
            const int   ti = bi[j]; bi[j] = bi[j - 1]; bi[j - 1] = ti;
          }
        }
      }
    }

    // Own-wave async copies done, then barrier => all staging visible and
    // dchunk free for the next compute phase.
    wait_async0();
    __syncthreads();
  }

  // ---- merge 32 lane-lists -> 24 global minima (wave min-pop) ----
  // key = (floatbits(dist) << 32) | index : non-negative floats order as
  // uints; low idx bits give top_k's first-occurrence tie-break.
  for (int r = 0; r < K_; ++r) {
    unsigned long long key =
        (((unsigned long long)__float_as_uint(bd[0])) << 32) | (unsigned)bi[0];
    unsigned long long m = key;
#pragma unroll
    for (int off = 16; off > 0; off >>= 1) {
      const unsigned long long o = __shfl_xor(m, (unsigned)off, 32);
      m = (o < m) ? o : m;
    }
    if (key == m) {  // winner lane pops (shift keeps static indexing)
#pragma unroll
      for (int j = 0; j < K_ - 1; ++j) { bd[j] = bd[j + 1]; bi[j] = bi[j + 1]; }
      bd[K_ - 1] = 1e30f;
      bi[K_ - 1] = 0;
    }
    if (lane == 0) topk[((size_t)(b * G_ + gBase + wv)) * K_ + r] = (int)(m & 0xffffffffu);
  }
}

// ---------------------------------------------------------------------------
// Kernel 3a: zero the per-batch sum-of-squares accumulators.
// ---------------------------------------------------------------------------
__global__ void zero_kernel(float* __restrict__ sumsq) {
  if (threadIdx.x < B_) sumsq[threadIdx.x] = 0.0f;
}

// ---------------------------------------------------------------------------
// Kernel 3b: per-(b,g) mean over K for all 67 channels + batch sum of
// squared deviations (the global std with ddof=1; mean-of-diff is 0).
// ---------------------------------------------------------------------------
__global__ __launch_bounds__(128) void stats_kernel(const float* __restrict__ xyz,
                                                    const float* __restrict__ points,
                                                    const int* __restrict__ topk,
                                                    float* __restrict__ meanbuf,
                                                    float* __restrict__ sumsq) {
  const int bg = blockIdx.x;
  const int b  = bg / G_;
  const int ch = threadIdx.x;

  __shared__ int nidx[K_];
  if (threadIdx.x < K_) nidx[threadIdx.x] = topk[(size_t)bg * K_ + threadIdx.x];
  __syncthreads();

  float vals[K_];
  float s = 0.0f;
  if (ch < CH_) {
#pragma unroll
    for (int k = 0; k < K_; ++k) {
      const int n = nidx[k];
      const float v = (ch < C_) ? points[((size_t)b * N_ + n) * C_ + ch]
                                : xyz[((size_t)b * N_ + n) * 3 + (ch - C_)];
      vals[k] = v;
      s += v;
    }
  }
  const float mu = s * (1.0f / (float)K_);
  float ss = 0.0f;
  if (ch < CH_) {
    meanbuf[(size_t)bg * CH_ + ch] = mu;
#pragma unroll
    for (int k = 0; k < K_; ++k) {
      const float d = vals[k] - mu;
      ss += d * d;
    }
  }
  // block reduce (4 waves of 32)
#pragma unroll
  for (int off = 16; off > 0; off >>= 1) ss += __shfl_down(ss, (unsigned)off, 32);
  __shared__ float ws[4];
  if ((threadIdx.x & 31) == 0) ws[threadIdx.x >> 5] = ss;
  __syncthreads();
  if (threadIdx.x == 0) {
    atomicAdd(&sumsq[b], ws[0] + ws[1] + ws[2] + ws[3]);
  }
}

// ---------------------------------------------------------------------------
// Kernel 4: produce new_points_out (B,G,K,131): normalized grouped channels
// (affine-transformed) followed by the broadcast sampled-point features.
// ---------------------------------------------------------------------------
__global__ __launch_bounds__(256) void finalize_kernel(const float* __restrict__ xyz,
                                                       const float* __restrict__ points,
                                                       const float* __restrict__ alpha,
                                                       const float* __restrict__ beta,
                                                       const int* __restrict__ fps_idx,
                                                       const int* __restrict__ topk,
                                                       const float* __restrict__ meanbuf,
                                                       const float* __restrict__ sumsq,
                                                       float* __restrict__ out2) {
  const int i = blockIdx.x * 256 + threadIdx.x;
  const int total = B_ * G_ * K_ * OC_;
  if (i >= total) return;

  const int ch = i % OC_;
  const int r  = i / OC_;
  const int k  = r % K_;
  const int g  = (r / K_) % G_;
  const int b  = r / (K_ * G_);

  float v;
  if (ch < CH_) {
    const int n = topk[((size_t)(b * G_ + g)) * K_ + k];
    const float raw = (ch < C_) ? points[((size_t)b * N_ + n) * C_ + ch]
                                : xyz[((size_t)b * N_ + n) * 3 + (ch - C_)];
    const float mu = meanbuf[((size_t)(b * G_ + g)) * CH_ + ch];
    const float M  = (float)(G_ * K_ * CH_) - 1.0f;
    const float sd = sqrtf(sumsq[b] / M);
    v = alpha[ch] * ((raw - mu) / (sd + 1e-5f)) + beta[ch];
  } else {
    const int n = fps_idx[b * G_ + g];
    v = points[((size_t)b * N_ + n) * C_ + (ch - CH_)];
  }
  out2[i] = v;
}

// ---------------------------------------------------------------------------
extern "C" void kernel_launch(void* const* d_in, const int* in_sizes, int n_in,
                              void* d_out, int out_size, void* d_ws, size_t ws_size,
                              hipStream_t stream) {
  const float* xyz    = (const float*)d_in[0];
  const float* points = (const float*)d_in[1];
  const float* alpha  = (const float*)d_in[2];
  const float* beta   = (const float*)d_in[3];
  // d_in[4] (groups) / d_in[5] (kneighbors) are device scalars; values are
  // fixed by the reference (2048 / 24) and baked in as compile-time consts.
  (void)in_sizes; (void)n_in; (void)out_size; (void)ws_size;

  float* new_xyz = (float*)d_out;                        // (B,G,3)
  float* out2    = (float*)d_out + (size_t)B_ * G_ * 3;  // (B,G,K,131)

  // Workspace layout
  int*   fps_idx = (int*)d_ws;                             // B*G ints
  int*   topk    = fps_idx + B_ * G_;                      // B*G*K ints
  float* meanbuf = (float*)(topk + (size_t)B_ * G_ * K_);  // B*G*67 floats
  float* sumsq   = meanbuf + (size_t)B_ * G_ * CH_;        // B floats

  fps_kernel<<<B_, 1024, 0, stream>>>(xyz, new_xyz, fps_idx);
  knn_kernel<<<B_ * (G_ / 16), 512, 0, stream>>>(xyz, new_xyz, topk);
  zero_kernel<<<1, 32, 0, stream>>>(sumsq);
  stats_kernel<<<B_ * G_, 128, 0, stream>>>(xyz, points, topk, meanbuf, sumsq);
  const int total = B_ * G_ * K_ * OC_;
  finalize_kernel<<<(total + 255) / 256, 256, 0, stream>>>(
      xyz, points, alpha, beta, fps_idx, topk, meanbuf, sumsq, out2);
}